// MultigridLayer_74783970558225
// MI455X (gfx1250) — compile-verified
//
#include <hip/hip_runtime.h>
#include <hip/hip_bf16.h>
#include <math.h>

typedef float v2f __attribute__((ext_vector_type(2)));
typedef float v4f __attribute__((ext_vector_type(4)));
typedef float v8f __attribute__((ext_vector_type(8)));

#define N_C   5120      // NUM_EQ + NUM_INEQ
#define N_EQ  1024
#define N_V   2048
#define ROW_OUT 13374   // per-batch output row length
// segment offsets within one output row
#define OFF_COEF 0
#define OFF_RHS  6144
#define OFF_S0   7168
#define OFF_S1   7199
#define OFF_D    7230
#define OFF_B    9278
#define OFF_X    11326
#define JW 0.3f

// ---------------------------------------------------------------------------
// Kernel 0: all the small downsampling outputs (bilinear align_corners=True,
// 64x64 -> 32x32, plus adjacent-pair step sums).
// ---------------------------------------------------------------------------
__global__ __launch_bounds__(256) void k_small(
    const float* __restrict__ coeffs, const float* __restrict__ rhs,
    const float* __restrict__ s0, const float* __restrict__ s1,
    float* __restrict__ out) {
  int tid = blockIdx.x * blockDim.x + threadIdx.x;
  const float scale = 63.0f / 31.0f;
  if (tid < 24576) {                         // new_coeffs: B*6*32*32
    int iw = tid & 31; int t = tid >> 5;
    int ih = t & 31;   t >>= 5;
    int o  = t % 6;    int b = t / 6;
    float sh = ih * scale; int h0 = (int)sh; if (h0 > 62) h0 = 62; float wh = sh - h0;
    float sw = iw * scale; int w0 = (int)sw; if (w0 > 62) w0 = 62; float ww = sw - w0;
    const float* cb = coeffs + (size_t)b * 4096 * 6;
#define CAT(h, w) cb[(size_t)((h) * 64 + (w)) * 6 + o]
    float v = (1.f - wh) * ((1.f - ww) * CAT(h0, w0)     + ww * CAT(h0, w0 + 1))
            +        wh  * ((1.f - ww) * CAT(h0 + 1, w0) + ww * CAT(h0 + 1, w0 + 1));
#undef CAT
    out[(size_t)b * ROW_OUT + OFF_COEF + (ih * 32 + iw) * 6 + o] = v;
  } else if (tid < 24576 + 4096) {           // new_rhs: B*32*32
    int t = tid - 24576;
    int iw = t & 31; int ih = (t >> 5) & 31; int b = t >> 10;
    float sh = ih * scale; int h0 = (int)sh; if (h0 > 62) h0 = 62; float wh = sh - h0;
    float sw = iw * scale; int w0 = (int)sw; if (w0 > 62) w0 = 62; float ww = sw - w0;
    const float* rb = rhs + (size_t)b * 4096;
#define RAT(h, w) rb[(h) * 64 + (w)]
    float v = (1.f - wh) * ((1.f - ww) * RAT(h0, w0)     + ww * RAT(h0, w0 + 1))
            +        wh  * ((1.f - ww) * RAT(h0 + 1, w0) + ww * RAT(h0 + 1, w0 + 1));
#undef RAT
    out[(size_t)b * ROW_OUT + OFF_RHS + ih * 32 + iw] = v;
  } else if (tid < 24576 + 4096 + 124) {     // new_steps0: B*31
    int t = tid - (24576 + 4096);
    int j = t % 31, b = t / 31;
    out[(size_t)b * ROW_OUT + OFF_S0 + j] = s0[b * 63 + 2 * j] + s0[b * 63 + 2 * j + 1];
  } else if (tid < 24576 + 4096 + 248) {     // new_steps1: B*31
    int t = tid - (24576 + 4096 + 124);
    int j = t % 31, b = t / 31;
    out[(size_t)b * ROW_OUT + OFF_S1 + j] = s1[b * 63 + 2 * j] + s1[b * 63 + 2 * j + 1];
  }
}

// ---------------------------------------------------------------------------
// Kernel 1: D = diag(At Pinv A), b = -At (Pinv * A_rhs). One coalesced
// streaming pass over A (168 MB total across grid).
// ---------------------------------------------------------------------------
__global__ __launch_bounds__(256) void k_diag(
    const float* __restrict__ A, const float* __restrict__ A_rhs,
    float* __restrict__ out) {
  int b = blockIdx.y;
  int v = blockIdx.x * 256 + threadIdx.x;
  const float* Ab = A + (size_t)b * N_C * N_V;
  const float* rb = A_rhs + (size_t)b * N_C;
  float d = 0.f, ar = 0.f;
  for (int c = 0; c < N_C; ++c) {
    float pinv = (c < N_EQ) ? 1.0f : 0.01f;   // 1/DS, DS = 100
    float a = Ab[(size_t)c * N_V + v];
    d  = fmaf(pinv * a, a, d);
    ar = fmaf(a, pinv * rb[c], ar);
  }
  out[(size_t)b * ROW_OUT + OFF_D + v] = d;
  out[(size_t)b * ROW_OUT + OFF_B + v] = -ar;
}

// ---------------------------------------------------------------------------
// Kernel 2: AtA[b] = A[b]^T * diag(Pinv) * A[b]  (2048x2048, K=5120) using
// V_WMMA_F32_16X16X4_F32. Workgroup = 128 threads (4 waves), 64x64 tile,
// each wave computes a 32x32 sub-tile with 2x2 WMMA accumulators.
//
// LDS tiles are stored k-innermost with padded stride 34 so each WMMA operand
// (two k-adjacent f32 per lane) is a single aligned ds_load_b64 straight into
// an even VGPR pair: no shuffle movs, no LDS bank conflicts on the read side.
// ---------------------------------------------------------------------------
#define TM 64
#define TN 64
#define KT 32
#define LDST 34   // padded k-stride in floats (even -> b64-aligned; kills 64-bank conflicts)

__global__ __launch_bounds__(128) void k_ata(
    const float* __restrict__ A, float* __restrict__ AtA) {
  int b  = blockIdx.y;
  int tm = blockIdx.x >> 5;          // /32
  int tn = blockIdx.x & 31;
  __shared__ __align__(16) float lsA[TM * LDST];   // [m][k]  (A^T operand tile)
  __shared__ __align__(16) float lsB[TN * LDST];   // [n][k]  (Pinv-scaled)
  const float* Ab = A + (size_t)b * N_C * N_V;
  int lane = threadIdx.x & 31;
  int wave = threadIdx.x >> 5;
  int wm = (wave >> 1) * 32;
  int wn = (wave & 1) * 32;
  int mBase = tm * TM, nBase = tn * TN;

  v8f acc00 = {}, acc01 = {}, acc10 = {}, acc11 = {};

  for (int c0 = 0; c0 < N_C; c0 += KT) {
    __syncthreads();   // previous-tile LDS reads done
#pragma unroll
    for (int i = 0; i < 4; ++i) {
      int idx = threadIdx.x + i * 128;     // 512 float4 per buffer
      int row = idx >> 4;                  // k row within tile
      int col = (idx & 15) << 2;           // m/n column (x4)
      const float* gsrc = Ab + (size_t)(c0 + row) * N_V;
      v4f va = *reinterpret_cast<const v4f*>(gsrc + mBase + col);
      lsA[(col + 0) * LDST + row] = va.x;
      lsA[(col + 1) * LDST + row] = va.y;
      lsA[(col + 2) * LDST + row] = va.z;
      lsA[(col + 3) * LDST + row] = va.w;
      float pinv = ((c0 + row) < N_EQ) ? 1.0f : 0.01f;
      v4f vb = *reinterpret_cast<const v4f*>(gsrc + nBase + col);
      vb = vb * pinv;
      lsB[(col + 0) * LDST + row] = vb.x;
      lsB[(col + 1) * LDST + row] = vb.y;
      lsB[(col + 2) * LDST + row] = vb.z;
      lsB[(col + 3) * LDST + row] = vb.w;
      if (c0 + KT < N_C)                    // pull next K-tile toward L2/L0
        __builtin_prefetch(gsrc + (size_t)KT * N_V + mBase + col, 0, 0);
    }
    __syncthreads();

    int l = lane & 15;
    int khalf = (lane >> 4) << 1;           // lanes 0-15 -> K pair (0,1); 16-31 -> (2,3)
#pragma unroll
    for (int k = 0; k < KT; k += 4) {
      v2f a0 = *reinterpret_cast<const v2f*>(&lsA[(wm + l)      * LDST + k + khalf]);
      v2f a1 = *reinterpret_cast<const v2f*>(&lsA[(wm + 16 + l) * LDST + k + khalf]);
      v2f b0 = *reinterpret_cast<const v2f*>(&lsB[(wn + l)      * LDST + k + khalf]);
      v2f b1 = *reinterpret_cast<const v2f*>(&lsB[(wn + 16 + l) * LDST + k + khalf]);
      acc00 = __builtin_amdgcn_wmma_f32_16x16x4_f32(false, a0, false, b0, (short)0, acc00, false, false);
      acc01 = __builtin_amdgcn_wmma_f32_16x16x4_f32(false, a0, false, b1, (short)0, acc01, false, false);
      acc10 = __builtin_amdgcn_wmma_f32_16x16x4_f32(false, a1, false, b0, (short)0, acc10, false, false);
      acc11 = __builtin_amdgcn_wmma_f32_16x16x4_f32(false, a1, false, b1, (short)0, acc11, false, false);
    }
  }

  // C/D layout: lanes 0-15 -> N = lane, M = r; lanes 16-31 -> N = lane-16, M = 8+r
  float* Cb = AtA + (size_t)b * N_V * N_V;
  int nCol = lane & 15;
  int mOff = (lane >> 4) * 8;
#pragma unroll
  for (int r = 0; r < 8; ++r) {
    size_t m0 = (size_t)(mBase + wm + mOff + r);
    size_t m1 = (size_t)(mBase + wm + 16 + mOff + r);
    Cb[m0 * N_V + nBase + wn + nCol]      = acc00[r];
    Cb[m0 * N_V + nBase + wn + 16 + nCol] = acc01[r];
    Cb[m1 * N_V + nBase + wn + nCol]      = acc10[r];
    Cb[m1 * N_V + nBase + wn + 16 + nCol] = acc11[r];
  }
}

// ---------------------------------------------------------------------------
// Kernel 3: one weighted-Jacobi step. AtA is symmetric, so we read its
// COLUMN v as row-major rows -> fully coalesced; AtA (67 MB) stays L2-resident.
// ---------------------------------------------------------------------------
__global__ __launch_bounds__(256) void k_jacobi(
    const float* __restrict__ AtA, const float* __restrict__ dseg,
    const float* __restrict__ xin, int inStride,
    float* __restrict__ xout, int outStride) {
  int b = blockIdx.y;
  int v = blockIdx.x * 256 + threadIdx.x;
  __shared__ float xs[N_V];
  const float* xb = xin + (size_t)b * inStride;
  for (int i = threadIdx.x; i < N_V; i += 256) xs[i] = xb[i];
  __syncthreads();

  const float* M = AtA + (size_t)b * N_V * N_V;
  float acc = 0.f;
  for (int r = 0; r < N_V; ++r) {
    if ((r & 63) == 0 && r + 64 < N_V)
      __builtin_prefetch(M + (size_t)(r + 64) * N_V + v, 0, 0);
    acc = fmaf(M[(size_t)r * N_V + v], xs[r], acc);   // (AtA x)[v] by symmetry
  }
  float D  = dseg[(size_t)b * ROW_OUT + OFF_D + v];
  float bb = dseg[(size_t)b * ROW_OUT + OFF_B + v];
  float x  = xs[v];
  xout[(size_t)b * outStride + v] = x + JW * (bb - acc) / D;
}

// ---------------------------------------------------------------------------
extern "C" void kernel_launch(void* const* d_in, const int* in_sizes, int n_in,
                              void* d_out, int out_size, void* d_ws, size_t ws_size,
                              hipStream_t stream) {
  const float* coeffs = (const float*)d_in[0];
  const float* rhs    = (const float*)d_in[1];
  const float* s0     = (const float*)d_in[2];
  const float* s1     = (const float*)d_in[3];
  const float* A      = (const float*)d_in[4];
  const float* A_rhs  = (const float*)d_in[5];
  const float* x0     = (const float*)d_in[6];
  float* out = (float*)d_out;

  // workspace: AtA (4*2048*2048 f32 = 64 MiB) + two x ping-pong buffers
  float* AtA   = (float*)d_ws;
  float* xping = AtA + (size_t)4 * N_V * N_V;
  float* xpong = xping + (size_t)4 * N_V;
  (void)in_sizes; (void)n_in; (void)out_size; (void)ws_size;

  // 0) small downsampling outputs
  k_small<<<(24576 + 4096 + 248 + 255) / 256, 256, 0, stream>>>(coeffs, rhs, s0, s1, out);

  // 1) D and AtPrhs (written straight into their output segments)
  k_diag<<<dim3(N_V / 256, 4), 256, 0, stream>>>(A, A_rhs, out);

  // 2) AtA GEMM via v_wmma_f32_16x16x4_f32
  k_ata<<<dim3((N_V / TM) * (N_V / TN), 4), 128, 0, stream>>>(A, AtA);

  // 3) 20 Jacobi steps; last one writes the strided x_sm output segment
  const float* xin = x0; int inS = N_V;
  float* bufs[2] = { xping, xpong };
  for (int it = 0; it < 20; ++it) {
    float* xo; int oS;
    if (it == 19) { xo = out + OFF_X; oS = ROW_OUT; }
    else          { xo = bufs[it & 1]; oS = N_V; }
    k_jacobi<<<dim3(N_V / 256, 4), 256, 0, stream>>>(AtA, out, xin, inS, xo, oS);
    xin = xo; inS = oS;
  }
}